// AttentionPool_18940805775523
// MI455X (gfx1250) — compile-verified
//
#include <hip/hip_runtime.h>

// ---------------------------------------------------------------------------
// Masked attention pooling, single-pass online softmax, split over N.
//   B=32, N=8192, D=256 fp32.  x read exactly once from HBM:
//   256 MB / 23.3 TB/s ~= 11 us floor -> bandwidth bound, so the kernel is
//   organized around data movement:
//     * global -> LDS staging uses GLOBAL_LOAD_ASYNC_TO_LDS_B128 (ASYNCcnt),
//       double-buffered so tile t+1 streams in while tile t is consumed.
//     * logits (x . q) run on the matrix pipe: V_WMMA_F32_16X16X4_F32 with q
//       replicated across the 16 B-columns; 8 waves split 4 position-groups
//       x 2 K-halves, partial logits summed in LDS.
//     * weighted accumulation is a bank-conflict-free column-parallel LDS
//       reduction (thread t owns output column d=t).
// ---------------------------------------------------------------------------

#define BATCH   32
#define NPOS    8192
#define DDIM    256
#define SPLITS  16
#define CHUNK   (NPOS / SPLITS)     // 512 positions per workgroup
#define TILE    64                  // positions per LDS tile (64 KB / buffer)
#define NTILES  (CHUNK / TILE)      // 8
#define THREADS 256                 // 8 wave32s
#define LOADS_PER_THREAD (TILE * DDIM * 4 / 16 / THREADS)   // 16 x b128

typedef float v2f __attribute__((ext_vector_type(2)));
typedef float v8f __attribute__((ext_vector_type(8)));

// Partial results in workspace: [B][SPLITS][2 + D]  (m, s, acc[256])
#define PART_STRIDE (DDIM + 2)

__global__ __launch_bounds__(THREADS)
void ap_partial_kernel(const float* __restrict__ x,
                       const unsigned char* __restrict__ mask,
                       const float* __restrict__ q,
                       float* __restrict__ part) {
    __shared__ float sm_x[2][TILE * DDIM];  // 2 x 64 KB double-buffered x tile
    __shared__ float sm_q[DDIM];            // 1 KB query
    __shared__ float sm_part[2][TILE];      // per-K-half partial logits
    __shared__ float sm_w[TILE];
    __shared__ float sm_mask[TILE];

    const int tid   = threadIdx.x;
    const int lane  = tid & 31;
    const int wave  = tid >> 5;
    const int split = blockIdx.x;
    const int b     = blockIdx.y;

    if (tid < DDIM) sm_q[tid] = q[tid];

    const float* xb = x + (size_t)b * NPOS * DDIM;
    const int chunk0 = split * CHUNK;

    // LDS byte offsets of the two buffers (low 32 bits of the generic pointer)
    const unsigned lds0 = (unsigned)(uintptr_t)(void*)&sm_x[0][0];
    const unsigned lds1 = (unsigned)(uintptr_t)(void*)&sm_x[1][0];

    float acc  = 0.0f;       // this thread's output column d = tid
    float runm = -3.0e38f;   // running max (finite sentinel avoids inf-inf NaN)
    float runs = 0.0f;       // running weight sum (identical across threads)

    // WMMA fp32 A-layout (16x4): lanes 0-15 hold K{0,1}, lanes 16-31 K{2,3}
    const int mrow  = lane & 15;
    const int khalf = lane >> 4;
    const int grp   = wave & 3;   // position group: 16 rows
    const int half  = wave >> 2;  // K half: [half*128, half*128+128)

    // ---- async-issue one tile (b128 per lane, ASYNCcnt-tracked) ----
    auto issue_tile = [&](int t) {
        const unsigned lbase = (t & 1) ? lds1 : lds0;
        const unsigned long long gbase =
            (unsigned long long)(uintptr_t)(xb + (size_t)(chunk0 + t * TILE) * DDIM);
#pragma unroll
        for (int i = 0; i < LOADS_PER_THREAD; ++i) {
            const unsigned idx = (unsigned)(i * THREADS + tid);
            unsigned           lofs = lbase + idx * 16u;
            unsigned long long gadr = gbase + (unsigned long long)idx * 16ull;
            asm volatile("global_load_async_to_lds_b128 %0, %1, off"
                         :: "v"(lofs), "v"(gadr) : "memory");
        }
    };

    issue_tile(0);

    for (int t = 0; t < NTILES; ++t) {
        const int base = chunk0 + t * TILE;

        // tile t resident + all waves done reading the buffer tile t+1 targets
        asm volatile("s_wait_asynccnt 0x0" ::: "memory");
        __syncthreads();

        if (t + 1 < NTILES) issue_tile(t + 1);     // overlap with phases A/B
        if (tid < TILE)
            sm_mask[tid] = mask[(size_t)b * NPOS + base + tid] ? 1.0f : 0.0f;

        const float* xt = sm_x[t & 1];

        // ---- phase A: logits on the matrix pipe ----
        // wave = (half, grp): 16 positions x 128 of K per wave.
        {
            v8f c = {};
            const float* arow = &xt[(grp * 16 + mrow) * DDIM + half * 128 + 2 * khalf];
            const float* brow = &sm_q[half * 128 + 2 * khalf];
#pragma unroll 4
            for (int k = 0; k < 128; k += 4) {
                v2f a  = *(const v2f*)(arow + k);   // A: x[row][k..k+1] per layout
                v2f bq = *(const v2f*)(brow + k);   // B: q replicated over 16 cols
                c = __builtin_amdgcn_wmma_f32_16x16x4_f32(
                        false, a, false, bq, (short)0, c, false, false);
            }
            // D: VGPR r, lanes 0-15 -> M=r, lanes 16-31 -> M=8+r (cols identical)
            if (lane == 0 || lane == 16) {
                const int nb = grp * 16 + (lane >> 1);
#pragma unroll
                for (int r = 0; r < 8; ++r) sm_part[half][nb + r] = c[r];
            }
        }
        __syncthreads();

        // ---- online softmax update (broadcast LDS reads, all threads) ----
        float tmax = -3.0e38f;
#pragma unroll 8
        for (int n = 0; n < TILE; ++n) {
            float l = sm_part[0][n] + sm_part[1][n];
            l = (sm_mask[n] > 0.0f) ? l : -3.0e38f;
            tmax = fmaxf(tmax, l);
        }
        const float newm  = fmaxf(runm, tmax);
        const float scale = __expf(runm - newm);
        acc  *= scale;
        runs *= scale;
        runm  = newm;
        if (tid < TILE) {
            const float l = sm_part[0][tid] + sm_part[1][tid];
            sm_w[tid] = (sm_mask[tid] > 0.0f) ? __expf(l - newm) : 0.0f;
        }
        __syncthreads();

        // ---- phase B: acc[d=tid] += sum_n w[n] * x[n][tid] ----
        float ssum = 0.0f;
#pragma unroll 4
        for (int n = 0; n < TILE; ++n) {
            const float w = sm_w[n];               // broadcast read
            ssum += w;
            acc  += w * xt[n * DDIM + tid];        // consecutive banks, no conflict
        }
        runs += ssum;
    }

    float* p = part + ((size_t)b * SPLITS + split) * PART_STRIDE;
    if (tid == 0) { p[0] = runm; p[1] = runs; }
    p[2 + tid] = acc;
}

__global__ __launch_bounds__(THREADS)
void ap_combine_kernel(const float* __restrict__ part, float* __restrict__ out) {
    const int b   = blockIdx.x;
    const int tid = threadIdx.x;   // d index, 256 == D

    float M = -3.0e38f;
#pragma unroll
    for (int s = 0; s < SPLITS; ++s)
        M = fmaxf(M, part[((size_t)b * SPLITS + s) * PART_STRIDE]);

    float tot = 0.0f, o = 0.0f;
#pragma unroll
    for (int s = 0; s < SPLITS; ++s) {
        const float* p = part + ((size_t)b * SPLITS + s) * PART_STRIDE;
        const float e = __expf(p[0] - M);
        tot += e * p[1];
        o   += e * p[2 + tid];
    }
    out[b * DDIM + tid] = o / tot;
}

extern "C" void kernel_launch(void* const* d_in, const int* in_sizes, int n_in,
                              void* d_out, int out_size, void* d_ws, size_t ws_size,
                              hipStream_t stream) {
    const float*         x    = (const float*)d_in[0];          // [B,N,D] fp32
    const unsigned char* mask = (const unsigned char*)d_in[1];  // [B,N] bool (1B)
    const float*         q    = (const float*)d_in[2];          // [D] fp32
    float*               out  = (float*)d_out;                  // [B,D] fp32
    float*               part = (float*)d_ws;                   // 32*16*258*4 = 516 KB

    (void)in_sizes; (void)n_in; (void)out_size; (void)ws_size;

    dim3 grid1(SPLITS, BATCH);
    ap_partial_kernel<<<grid1, THREADS, 0, stream>>>(x, mask, q, part);
    ap_combine_kernel<<<BATCH, THREADS, 0, stream>>>(part, out);
}